// Model_74663711473754
// MI455X (gfx1250) — compile-verified
//
#include <hip/hip_runtime.h>
#include <hip/hip_bf16.h>

// Problem constants (from reference): B=128, S=4096, C=64, P=24, STEPS=96, D=1.
// NSER = B*C = 8192 independent AR(24) series.
#define NSER   8192
#define PW     24
#define NSTEP  96

typedef __attribute__((ext_vector_type(2))) float v2f;
typedef __attribute__((ext_vector_type(8))) float v8f;

// ---------------------------------------------------------------------------
// Kernel 1: build the 8192x24 window matrix (K-major: W0k[k][series]) and the
// per-series last observed value. Fully coalesced: consecutive threads map to
// consecutive series -> consecutive channels in x_enc.
// 800 blocks * 256 threads == 25*8192 threads exactly (24 window cols + xlast).
// ---------------------------------------------------------------------------
__global__ void arima_diff_kernel(const float* __restrict__ x,
                                  float* __restrict__ W0k,
                                  float* __restrict__ xlast) {
    int idx = blockIdx.x * 256 + threadIdx.x;   // [0, 25*8192)
    int j = idx >> 13;                          // 0..24
    int s = idx & (NSER - 1);                   // series = b*64 + c
    int b = s >> 6;
    int c = s & 63;
    if (j < PW) {
        // diff[b, 4071+j, c] = x[b, 4072+j, c] - x[b, 4071+j, c]
        long base = ((long)(b * 4096 + 4071 + j)) * 64 + c;
        W0k[j * NSER + s] = x[base + 64] - x[base];
    } else {
        xlast[s] = x[((long)(b * 4096 + 4095)) * 64 + c];
    }
}

// ---------------------------------------------------------------------------
// Kernel 2 (single wave): companion-matrix recurrence producing the 96x24
// cumulative operator cT[t][k] = sum_{s<=t} row_s[k] and cumulative constants
// cK[t] = sum_{s<=t} k_s.  Lane e owns column e of the 24x24 window-of-rows
// (circular buffer in LDS), so all LDS deps are same-lane; single-wave
// workgroup => barriers lower to S_NOP, hardware keeps DS ops in order.
// ---------------------------------------------------------------------------
__global__ void arima_coeff_kernel(const float* __restrict__ arw,
                                   const float* __restrict__ arb,
                                   const float* __restrict__ mab,
                                   float* __restrict__ cT,
                                   float* __restrict__ cK) {
    __shared__ float W[PW][PW];   // W[row][col]: coefficient rows of window
    __shared__ float Wk[PW];      // constant term of each window entry
    __shared__ float phi[PW];

    int lane = threadIdx.x;
    for (int i = lane; i < PW * PW; i += 32)
        W[i / PW][i % PW] = (i / PW == i % PW) ? 1.0f : 0.0f;
    if (lane < PW) { Wk[lane] = 0.0f; phi[lane] = arw[lane]; }
    __syncthreads();

    float bsum = arb[0] + mab[0];   // ma residual window is identically zero
    float cum = 0.0f, cumk = 0.0f;
    for (int t = 0; t < NSTEP; ++t) {
        float r = 0.0f, sk = bsum;
        for (int j = 0; j < PW; ++j) {
            int row = (t + j) % PW;           // circular window
            float p = phi[j];
            if (lane < PW) r += p * W[row][lane];
            sk += p * Wk[row];
        }
        __syncthreads();
        if (lane < PW) {
            W[t % PW][lane] = r;              // slot t%PW held the dropped entry
            cum += r;
            cT[t * PW + lane] = cum;          // fused cumsum
        }
        if (lane == 0) {
            Wk[t % PW] = sk;
            cumk += sk;
            cK[t] = cumk;
        }
        __syncthreads();
    }
}

// ---------------------------------------------------------------------------
// Kernel 3: the whole forecast as one fp32 WMMA GEMM.
//   D[t][series] = sum_k cT[t][k] * W0k[k][series]   (M=96, N=8192, K=24)
// One wave per 16x16 tile; K=24 -> 6 chunks of V_WMMA_F32_16X16X4_F32.
// Epilogue adds cK[t] + xlast[series] and stores directly into (B,96,C)
// layout; lanes map to contiguous channels -> coalesced stores.
// ---------------------------------------------------------------------------
__global__ void arima_wmma_gemm(const float* __restrict__ W0k,
                                const float* __restrict__ xlast,
                                const float* __restrict__ cT,
                                const float* __restrict__ cK,
                                float* __restrict__ out) {
    int gtid  = blockIdx.x * blockDim.x + threadIdx.x;
    int wave  = gtid >> 5;          // 3072 tiles total
    int lane  = gtid & 31;
    int tileT = wave % (NSTEP / 16);    // 6 step tiles
    int tileS = wave / (NSTEP / 16);    // 512 series tiles
    int laneM = lane & 15;
    int laneH = lane >> 4;

    int ta = tileT * 16 + laneM;        // A-matrix row (step t)
    int sb = tileS * 16 + laneM;        // B-matrix col (series)

    v8f acc = {};
#pragma unroll
    for (int kk = 0; kk < 6; ++kk) {
        int k0 = kk * 4 + laneH * 2;    // lane-half owns K and K+1 of {K..K+3}
        v2f a, b;
        a.x = cT[ta * PW + k0];         // A: 16x4 fp32 fragment (2 VGPRs)
        a.y = cT[ta * PW + k0 + 1];
        b.x = W0k[k0 * NSER + sb];      // B: 4x16 fp32 fragment (coalesced)
        b.y = W0k[(k0 + 1) * NSER + sb];
        acc = __builtin_amdgcn_wmma_f32_16x16x4_f32(
            /*neg_a=*/false, a, /*neg_b=*/false, b,
            /*c_mod=*/(short)0, acc, /*reuse_a=*/false, /*reuse_b=*/false);
    }

    float xl = xlast[sb];
    int bi = sb >> 6, ci = sb & 63;
#pragma unroll
    for (int v = 0; v < 8; ++v) {
        int t = tileT * 16 + laneH * 8 + v;   // D layout: M = vgpr + 8*lane-half
        out[(bi * NSTEP + t) * 64 + ci] = acc[v] + cK[t] + xl;
    }
}

extern "C" void kernel_launch(void* const* d_in, const int* in_sizes, int n_in,
                              void* d_out, int out_size, void* d_ws, size_t ws_size,
                              hipStream_t stream) {
    const float* x_enc = (const float*)d_in[0];   // (128, 4096, 64) fp32
    const float* ar_w  = (const float*)d_in[1];   // (24,)
    const float* ar_b  = (const float*)d_in[2];   // scalar
    // d_in[3] = ma_w: residual window is identically zero -> unused
    const float* ma_b  = (const float*)d_in[4];   // scalar
    // d_in[5] = steps (constant 96 in this problem)
    float* out = (float*)d_out;                   // (128, 96, 64) fp32

    // Workspace layout (floats): ~829 KB total
    float* W0k  = (float*)d_ws;            // 24 * 8192  (K-major window matrix)
    float* xl   = W0k + PW * NSER;         // 8192
    float* cT   = xl + NSER;               // 96 * 24    (cumulative operator)
    float* cK   = cT + NSTEP * PW;         // 96

    arima_diff_kernel<<<(25 * NSER) / 256, 256, 0, stream>>>(x_enc, W0k, xl);
    arima_coeff_kernel<<<1, 32, 0, stream>>>(ar_w, ar_b, ma_b, cT, cK);
    // 3072 tiles / 8 waves per block = 384 blocks
    arima_wmma_gemm<<<384, 256, 0, stream>>>(W0k, xl, cT, cK, out);
}